// GlobalAttentionPool_3934190044025
// MI455X (gfx1250) — compile-verified
//
#include <hip/hip_runtime.h>
#include <math.h>

#define D_DIM 256
#define NEG_SLOPE 0.2f

typedef __attribute__((ext_vector_type(2))) float v2f;
typedef __attribute__((ext_vector_type(8))) float v8f;

// ---------------- helpers ----------------

__device__ __forceinline__ float lrelu(float v) {
    return v >= 0.0f ? v : NEG_SLOPE * v;
}

// Branchless-correct float atomic max via int/uint ordering trick.
// Works for any mix of signs; init value must be a real float (-INF ok).
__device__ __forceinline__ void atomicMaxF(float* addr, float v) {
    if (v >= 0.0f) {
        atomicMax((int*)addr, __float_as_int(v));
    } else {
        atomicMin((unsigned int*)addr, __float_as_uint(v));
    }
}

// ---------------- K1: h = x @ W via V_WMMA_F32_16X16X4_F32 ----------------
// One wave (32 lanes) computes h for 16 nodes. A = 16x4 tile of x rows,
// B = W[k] replicated across all 16 columns so every D column equals h.
// A/B layout (ISA 7.12.2): lanes 0-15 index M (A) / N (B); K = comp + 2*(lane>=16).

__global__ void k_h_wmma(const float* __restrict__ x, const float* __restrict__ W,
                         float* __restrict__ h, int N) {
    __shared__ float Wl[D_DIM];
    const int tid = threadIdx.x;
    for (int i = tid; i < D_DIM; i += blockDim.x) Wl[i] = W[i];
    __syncthreads();

    const int lane = tid & 31;
    const int wave = tid >> 5;
    const int row  = lane & 15;
    const int half = lane >> 4;                       // 0 or 1
    const long nodeBase = ((long)blockIdx.x * 8 + wave) * 16;
    long node = nodeBase + row;
    if (node > (long)N - 1) node = (long)N - 1;       // clamp (keep EXEC all-1s)

    const float* xr = x + node * D_DIM + 2 * half;    // this lane's K offset
    const float* Wr = Wl + 2 * half;

    v8f c = {};
    for (int k0 = 0; k0 < D_DIM; k0 += 4) {
        v2f a, b;
        a.x = xr[k0];
        a.y = xr[k0 + 1];
        b.x = Wr[k0];
        b.y = Wr[k0 + 1];
        c = __builtin_amdgcn_wmma_f32_16x16x4_f32(false, a, false, b,
                                                  (short)0, c, false, false);
    }

    // D layout: lane L<16 vgpr v -> [M=v][Nc=L]; lane L>=16 vgpr v -> [M=v+8][Nc=L-16]
    // Every column identical; lane 0 stores rows 0..7, lane 16 stores rows 8..15.
    if (lane == 0) {
        for (int v = 0; v < 8; ++v) {
            long n2 = nodeBase + v;
            if (n2 < N) h[n2] = c[v];
        }
    } else if (lane == 16) {
        for (int v = 0; v < 8; ++v) {
            long n2 = nodeBase + 8 + v;
            if (n2 < N) h[n2] = c[v];
        }
    }
}

// ---------------- K2: self-loop seeds per-dst max ----------------
__global__ void k_self_max(const float* __restrict__ h, float* __restrict__ m,
                           const float* __restrict__ att_src,
                           const float* __restrict__ att_dst, int N) {
    int n = blockIdx.x * blockDim.x + threadIdx.x;
    if (n >= N) return;
    float coef = att_src[0] + att_dst[0];
    m[n] = lrelu(coef * h[n]);
}

// ---------------- K3: edge pass, max into m[dst] ----------------
__global__ void k_edge_max(const int* __restrict__ src, const int* __restrict__ dst,
                           const float* __restrict__ h, float* __restrict__ m,
                           const float* __restrict__ att_src,
                           const float* __restrict__ att_dst, int E) {
    int i = blockIdx.x * blockDim.x + threadIdx.x;
    if (i >= E) return;
    int s = src[i], d = dst[i];
    float e = lrelu(att_src[0] * h[s] + att_dst[0] * h[d]);
    atomicMaxF(&m[d], e);
}

// ---------------- K4: self-loop seeds exp-sum and weighted acc ----------------
__global__ void k_self_exp(const float* __restrict__ h, const float* __restrict__ m,
                           float* __restrict__ s, float* __restrict__ acc,
                           const float* __restrict__ att_src,
                           const float* __restrict__ att_dst, int N) {
    int n = blockIdx.x * blockDim.x + threadIdx.x;
    if (n >= N) return;
    float hv = h[n];
    float e  = lrelu((att_src[0] + att_dst[0]) * hv);
    float w  = expf(e - m[n]);
    s[n]   = w;
    acc[n] = w * hv;
}

// ---------------- K5: edge pass, accumulate exp and exp*h[src] ----------------
__global__ void k_edge_exp(const int* __restrict__ src, const int* __restrict__ dst,
                           const float* __restrict__ h, const float* __restrict__ m,
                           float* __restrict__ s, float* __restrict__ acc,
                           const float* __restrict__ att_src,
                           const float* __restrict__ att_dst, int E) {
    int i = blockIdx.x * blockDim.x + threadIdx.x;
    if (i >= E) return;
    int sn = src[i], d = dst[i];
    float hs = h[sn];
    float e  = lrelu(att_src[0] * hs + att_dst[0] * h[d]);
    float w  = expf(e - m[d]);
    atomicAdd(&s[d], w);
    atomicAdd(&acc[d], w * hs);
}

// ---------------- K6: x_conv = acc/(s+eps) + bias (written over h buffer) ----------------
__global__ void k_xconv(const float* __restrict__ s, const float* __restrict__ acc,
                        float* __restrict__ xconv, const float* __restrict__ bias, int N) {
    int n = blockIdx.x * blockDim.x + threadIdx.x;
    if (n >= N) return;
    xconv[n] = acc[n] / (s[n] + 1e-16f) + bias[0];
}

// ---------------- K7: init per-graph max/sum ----------------
__global__ void k_ginit(float* __restrict__ gmax, float* __restrict__ gsum, int G) {
    int g = blockIdx.x * blockDim.x + threadIdx.x;
    if (g >= G) return;
    gmax[g] = -INFINITY;
    gsum[g] = 0.0f;
}

// ---------------- K8: per-graph max of xconv ----------------
__global__ void k_gmax(const int* __restrict__ batch, const float* __restrict__ xconv,
                       float* __restrict__ gmax, int N) {
    int n = blockIdx.x * blockDim.x + threadIdx.x;
    if (n >= N) return;
    atomicMaxF(&gmax[batch[n]], xconv[n]);
}

// ---------------- K9: p = exp(xconv - gmax); gsum += p ----------------
__global__ void k_gexp(const int* __restrict__ batch, const float* __restrict__ xconv,
                       const float* __restrict__ gmax, float* __restrict__ p,
                       float* __restrict__ gsum, int N) {
    int n = blockIdx.x * blockDim.x + threadIdx.x;
    if (n >= N) return;
    int g = batch[n];
    float mg = gmax[g];
    if (!isfinite(mg)) mg = 0.0f;   // matches reference empty-segment guard
    float w = expf(xconv[n] - mg);
    p[n] = w;
    atomicAdd(&gsum[g], w);
}

// ---------------- K10: per-graph pooled sum, one block per graph ----------------
__global__ void k_pool(const float* __restrict__ x, const int* __restrict__ batch,
                       const float* __restrict__ p, const float* __restrict__ gsum,
                       float* __restrict__ out, int N) {
    const int g = blockIdx.x;
    // batch is sorted: binary search [start, end) of nodes in graph g
    int lo = 0, hi = N;
    while (lo < hi) { int mid = (lo + hi) >> 1; if (batch[mid] <  g)     lo = mid + 1; else hi = mid; }
    const int start = lo;
    lo = start; hi = N;
    while (lo < hi) { int mid = (lo + hi) >> 1; if (batch[mid] <  g + 1) lo = mid + 1; else hi = mid; }
    const int end = lo;

    const float denom = gsum[g] + 1e-16f;
    const int t = threadIdx.x;          // t owns output dim t (D_DIM == blockDim.x)
    float acc = 0.0f;
    for (int n = start; n < end; ++n) {
        float score = p[n] / denom;
        acc += score * x[(long)n * D_DIM + t];
    }
    out[(long)g * D_DIM + t] = acc;
}

// ---------------- launch ----------------
extern "C" void kernel_launch(void* const* d_in, const int* in_sizes, int n_in,
                              void* d_out, int out_size, void* d_ws, size_t ws_size,
                              hipStream_t stream) {
    const float* x        = (const float*)d_in[0];
    const int*   ei       = (const int*)  d_in[1];
    const int*   batch    = (const int*)  d_in[2];
    const float* W        = (const float*)d_in[3];
    const float* att_src  = (const float*)d_in[4];
    const float* att_dst  = (const float*)d_in[5];
    const float* bias     = (const float*)d_in[6];
    float* out = (float*)d_out;

    const int N = in_sizes[0] / D_DIM;
    const int E = in_sizes[1] / 2;
    const int G = out_size / D_DIM;
    const int* src = ei;
    const int* dst = ei + E;

    // workspace layout (floats): h/xconv | m/p | s | acc | gmax | gsum
    float* f    = (float*)d_ws;
    float* h    = f;                 // N  (reused as xconv)
    float* m    = f + (size_t)N;     // N  (reused as p)
    float* sum  = f + (size_t)2 * N; // N
    float* acc  = f + (size_t)3 * N; // N
    float* gmax = f + (size_t)4 * N;         // G
    float* gsum = f + (size_t)4 * N + G;     // G
    float* xconv = h;
    float* p     = m;

    const int TB = 256;
    const int nodeBlocks = (N + TB - 1) / TB;
    const int edgeBlocks = (E + TB - 1) / TB;
    const int gBlocks    = (G + TB - 1) / TB;
    const int wmmaBlocks = (N + 127) / 128;  // 8 waves/block * 16 nodes/wave

    k_h_wmma  <<<wmmaBlocks, TB, 0, stream>>>(x, W, h, N);
    k_self_max<<<nodeBlocks, TB, 0, stream>>>(h, m, att_src, att_dst, N);
    k_edge_max<<<edgeBlocks, TB, 0, stream>>>(src, dst, h, m, att_src, att_dst, E);
    k_self_exp<<<nodeBlocks, TB, 0, stream>>>(h, m, sum, acc, att_src, att_dst, N);
    k_edge_exp<<<edgeBlocks, TB, 0, stream>>>(src, dst, h, m, sum, acc, att_src, att_dst, E);
    k_xconv   <<<nodeBlocks, TB, 0, stream>>>(sum, acc, xconv, bias, N);
    k_ginit   <<<gBlocks,    TB, 0, stream>>>(gmax, gsum, G);
    k_gmax    <<<nodeBlocks, TB, 0, stream>>>(batch, xconv, gmax, N);
    k_gexp    <<<nodeBlocks, TB, 0, stream>>>(batch, xconv, gmax, p, gsum, N);
    k_pool    <<<G,          D_DIM, 0, stream>>>(x, batch, p, gsum, out, N);
}